// DynamicRouting_111669150355
// MI455X (gfx1250) — compile-verified
//
#include <hip/hip_runtime.h>
#include <math.h>

#define B_ 64
#define I_ 1152
#define O_ 32
#define D_ 8
#define H_ 16

typedef float v2f __attribute__((ext_vector_type(2)));
typedef float v4f __attribute__((ext_vector_type(4)));
typedef float v8f __attribute__((ext_vector_type(8)));

// -----------------------------------------------------------------------------
// Kernel A: u_hat[b,i,o,h] = sum_d u[b,i,d] * W[i,o,d,h]  via V_WMMA_F32_16X16X4_F32
// Per i this is C[64,512] = A[64,8] x B[8,512]; an N-tile of 16 == one 'o', so the
// B tile W[i,o,:,:] is a contiguous 8x16 fp32 block. Each wave owns one 16x16
// (b-tile, o) output tile and loops over 18 i's; the workgroup (8 waves) covers
// 144 consecutive i's and emits one partial of s0 = sum_i u_hat (uniform-coupling
// iteration 0 is fused here for free).
// grid: 1024 = chunk(8) * o(32) * m0(4); block: 256 (8 waves)
// -----------------------------------------------------------------------------
__global__ __launch_bounds__(256) void uhat_wmma_kernel(
    const float* __restrict__ u, const float* __restrict__ w,
    float* __restrict__ uhat, float* __restrict__ psum)
{
    const int tid   = threadIdx.x;
    const int wave  = tid >> 5;
    const int lane  = tid & 31;
    const int bx    = blockIdx.x;
    const int chunk = bx >> 7;        // 0..7   (144 i's each)
    const int rem   = bx & 127;
    const int o     = rem >> 2;       // 0..31
    const int m0    = rem & 3;        // 0..3   (b tile base = m0*16)

    const int mloc = lane & 15;
    const int hi8  = (lane < 16) ? 0 : 8;   // C/D: lanes 16-31 hold M+8 rows
    const int koff = (lane < 16) ? 0 : 2;   // A/B f32 K striping across half-waves

    const float* uRow  = u + (size_t)(m0 * 16 + mloc) * (I_ * D_);
    const float* wBase = w + (size_t)o * (D_ * H_);

    v8f acc = {0.f,0.f,0.f,0.f,0.f,0.f,0.f,0.f};

    const int iBase = chunk * 144 + wave * 18;
    for (int ii = 0; ii < 18; ++ii) {
        const int i = iBase + ii;
        const float* uP = uRow  + i * D_;
        const float* wP = wBase + (size_t)i * (O_ * D_ * H_);

        v8f c = {0.f,0.f,0.f,0.f,0.f,0.f,0.f,0.f};
#pragma unroll
        for (int k0 = 0; k0 < D_; k0 += 4) {
            v2f a, b;
            a.x = uP[k0 + koff];                       // A vgpr0: K = k0+koff
            a.y = uP[k0 + koff + 1];                   // A vgpr1: K = k0+koff+1
            b.x = wP[(k0 + koff)     * H_ + mloc];     // B vgpr0: row K, N = lane%16
            b.y = wP[(k0 + koff + 1) * H_ + mloc];     // B vgpr1
            c = __builtin_amdgcn_wmma_f32_16x16x4_f32(
                    false, a, false, b, (short)0, c, false, false);
        }

        // store this i's u_hat tile and fold into the s0 partial
        float* outP = uhat + (size_t)i * (O_ * H_) + (size_t)o * H_ + mloc;
#pragma unroll
        for (int r = 0; r < 8; ++r) {
            const int bIdx = m0 * 16 + r + hi8;
            outP[(size_t)bIdx * ((size_t)I_ * O_ * H_)] = c[r];
            acc[r] += c[r];
        }
    }

    // cross-wave reduction of the s0 partial (sum over this block's 144 i's)
    __shared__ float red[8][32][8];
#pragma unroll
    for (int r = 0; r < 8; ++r) red[wave][lane][r] = acc[r];
    __syncthreads();
    if (wave == 0) {
#pragma unroll
        for (int r = 0; r < 8; ++r) {
            float s = 0.f;
#pragma unroll
            for (int wv = 0; wv < 8; ++wv) s += red[wv][lane][r];
            const int bIdx = m0 * 16 + r + hi8;
            psum[(((size_t)chunk * B_ + bIdx) * O_ + o) * H_ + mloc] = s;
        }
    }
}

// -----------------------------------------------------------------------------
// Kernel B: one routing sweep over u_hat. Lane == o (32 lanes = 32 output caps).
// logits[b,i,o] = sum_h vsum[b,o,h]*u_hat[b,i,o,h]; softmax over o is a 32-lane
// shuffle reduction; partial s[b,o,h] accumulated in registers over a 36-i chunk.
// grid: 256 blocks * 8 waves = 2048 wave-tasks = b(64) * chunk(32)
// -----------------------------------------------------------------------------
__global__ __launch_bounds__(256) void route_kernel(
    const float* __restrict__ uhat, const float* __restrict__ vsum,
    float* __restrict__ psum)
{
    const int tid   = threadIdx.x;
    const int wave  = tid >> 5;
    const int lane  = tid & 31;                 // lane == o
    const int task  = blockIdx.x * 8 + wave;
    const int b     = task >> 5;                // 0..63
    const int chunk = task & 31;                // 0..31

    float vs[16];
    {
        const float* vP = vsum + ((size_t)b * O_ + lane) * H_;
#pragma unroll
        for (int j = 0; j < 4; ++j) {
            v4f t = *(const v4f*)(vP + j * 4);
            vs[j*4+0] = t.x; vs[j*4+1] = t.y; vs[j*4+2] = t.z; vs[j*4+3] = t.w;
        }
    }

    float acc[16];
#pragma unroll
    for (int h = 0; h < 16; ++h) acc[h] = 0.f;

    const int iBase = chunk * 36;
    for (int ii = 0; ii < 36; ++ii) {
        const int i = iBase + ii;
        const float* uP = uhat + (((size_t)b * I_ + i) * O_ + lane) * H_;
        float uh[16];
#pragma unroll
        for (int j = 0; j < 4; ++j) {
            v4f t = *(const v4f*)(uP + j * 4);
            uh[j*4+0] = t.x; uh[j*4+1] = t.y; uh[j*4+2] = t.z; uh[j*4+3] = t.w;
        }
        float logit = 0.f;
#pragma unroll
        for (int h = 0; h < 16; ++h) logit = fmaf(vs[h], uh[h], logit);

        // softmax over o (32 lanes)
        float m = logit;
#pragma unroll
        for (int d = 16; d >= 1; d >>= 1) m = fmaxf(m, __shfl_xor(m, d, 32));
        const float e = __expf(logit - m);
        float ssum = e;
#pragma unroll
        for (int d = 16; d >= 1; d >>= 1) ssum += __shfl_xor(ssum, d, 32);
        const float cpl = e / ssum;

#pragma unroll
        for (int h = 0; h < 16; ++h) acc[h] = fmaf(cpl, uh[h], acc[h]);
    }

    float* pP = psum + (((size_t)chunk * B_ + b) * O_ + lane) * H_;
#pragma unroll
    for (int j = 0; j < 4; ++j) {
        v4f t; t.x = acc[j*4]; t.y = acc[j*4+1]; t.z = acc[j*4+2]; t.w = acc[j*4+3];
        *(v4f*)(pP + j * 4) = t;
    }
}

// -----------------------------------------------------------------------------
// Kernel R: s = scale * sum_p partial[p]; v = squash(s); vsum (+)= v; out = v.
// Threads map t -> (b,o,h) with h = t%16, so the 16-lane sum_h s^2 for squash is
// 4 xor-shuffle steps within aligned 16-lane half-wave groups.
// grid: 128, block: 256 (32768 threads = 64*32*16)
// -----------------------------------------------------------------------------
__global__ __launch_bounds__(256) void reduce_squash_kernel(
    const float* __restrict__ psum, int np, float scale, int addToVsum,
    float* __restrict__ vsum, float* __restrict__ out)
{
    const int t = blockIdx.x * 256 + threadIdx.x;    // 0..32767
    float s = 0.f;
    for (int p = 0; p < np; ++p) s += psum[(size_t)p * (B_ * O_ * H_) + t];
    s *= scale;

    float sq = s * s;
#pragma unroll
    for (int d = 8; d >= 1; d >>= 1) sq += __shfl_xor(sq, d, 32);

    const float v = (sq / (1.f + sq)) * s * rsqrtf(sq + 1e-8f);
    const float vnew = addToVsum ? (vsum[t] + v) : v;
    vsum[t] = vnew;
    out[t]  = v;
}

// -----------------------------------------------------------------------------
extern "C" void kernel_launch(void* const* d_in, const int* in_sizes, int n_in,
                              void* d_out, int out_size, void* d_ws, size_t ws_size,
                              hipStream_t stream) {
    const float* u = (const float*)d_in[0];   // [64,1152,8]
    const float* w = (const float*)d_in[1];   // [1152,32,8,16]
    float* out = (float*)d_out;               // [64,32,16]

    float* ws   = (float*)d_ws;
    float* uhat = ws;                                         // 37,748,736 floats (151 MB)
    float* psum = uhat + (size_t)B_ * I_ * O_ * H_;           //  1,048,576 floats (4 MB)
    float* vsum = psum + (size_t)32 * B_ * O_ * H_;           //     32,768 floats

    // iteration 0 fused into u_hat production (uniform coupling = 1/32)
    uhat_wmma_kernel<<<1024, 256, 0, stream>>>(u, w, uhat, psum);
    reduce_squash_kernel<<<128, 256, 0, stream>>>(psum, 8, 1.0f / 32.0f, 0, vsum, out);

    // iteration 1: logits_1 = <v0, u_hat>
    route_kernel<<<256, 256, 0, stream>>>(uhat, vsum, psum);
    reduce_squash_kernel<<<128, 256, 0, stream>>>(psum, 32, 1.0f, 1, vsum, out);

    // iteration 2: logits_2 = <v0+v1, u_hat>  (logits are linear in the v's)
    route_kernel<<<256, 256, 0, stream>>>(uhat, vsum, psum);
    reduce_squash_kernel<<<128, 256, 0, stream>>>(psum, 32, 1.0f, 1, vsum, out);
}